// CGCN_50165218017366
// MI455X (gfx1250) — compile-verified
//
#include <hip/hip_runtime.h>
#include <hip/hip_bf16.h>

// ---------------- problem constants ----------------
#define NUSER 60000
#define NITEM 40000
#define NN    100000            // NUSER + NITEM
#define DIMC  64
#define NE    1000000
#define NE2   2000000           // symmetrized edges
#define NW    (NN * DIMC)       // 6,400,000 floats
#define EPS_F 1e-16f
#define THR_F 0.02f
#define SLOPE 0.2f
#define INV_SQRT_D 0.125f

typedef __attribute__((ext_vector_type(2))) float v2f;
typedef __attribute__((ext_vector_type(8))) float v8f;

// order-preserving float <-> u32 map for atomic max
__device__ __forceinline__ unsigned f2ord(float f) {
    unsigned b = __float_as_uint(f);
    return (b & 0x80000000u) ? ~b : (b | 0x80000000u);
}
__device__ __forceinline__ float ord2f(unsigned u) {
    return (u & 0x80000000u) ? __uint_as_float(u & 0x7fffffffu)
                             : __uint_as_float(~u);
}
#define ORD_MIN 0x007FFFFFu   // f2ord(-inf)

// symmetrized edge list: e<NE: src=users[e], dst=items[e]; else swapped
__device__ __forceinline__ void edge_sd(const int* __restrict__ ei0, int e,
                                        int& s, int& d) {
    const int* ei1 = ei0 + NE;
    if (e < NE) { s = ei0[e]; d = ei1[e]; }
    else        { s = ei1[e - NE]; d = ei0[e - NE]; }
}

// per-wave L2 normalize of a 64-wide row; lane holds 2 components
__device__ __forceinline__ float2 l2pair(float ax, float ay) {
    float ss = ax * ax + ay * ay;
    #pragma unroll
    for (int m = 16; m; m >>= 1) ss += __shfl_xor(ss, m, 32);
    float inv = 1.0f / fmaxf(sqrtf(ss), 1e-12f);
    float2 o; o.x = ax * inv; o.y = ay * inv; return o;
}

// ---------------- kernels ----------------

// build x (= d_out head), y, z from normalized inputs. One wave per row.
__global__ void k_init_nodes(const float* __restrict__ f1,
                             const float* __restrict__ f2,
                             const float* __restrict__ f3,
                             const float* __restrict__ pref,
                             float* __restrict__ x,
                             float* __restrict__ y,
                             float* __restrict__ z) {
    int row = blockIdx.x * (blockDim.x >> 5) + (threadIdx.x >> 5);
    if (row >= NN) return;
    int lane = threadIdx.x & 31;
    unsigned col = lane * 2;
    unsigned ob  = (unsigned)row * DIMC + col;
    if (row < NUSER) {
        float2 v = *(const float2*)(pref + ob);
        float2 o = l2pair(v.x, v.y);
        *(float2*)(x + ob) = o;
        *(float2*)(y + ob) = o;
        *(float2*)(z + ob) = o;
    } else {
        unsigned ib = (unsigned)(row - NUSER) * DIMC + col;
        float2 v1 = *(const float2*)(f1 + ib);
        float2 o1 = l2pair(v1.x, v1.y);
        *(float2*)(x + ob) = o1;
        float2 v2 = *(const float2*)(f2 + ib);
        float2 o2 = l2pair(v2.x, v2.y);
        *(float2*)(y + ob) = o2;
        float2 v3 = *(const float2*)(f3 + ib);
        float2 o3 = l2pair(v3.x, v3.y);
        *(float2*)(z + ob) = o3;
    }
}

// zero x_hat; reset segment-max / denominators
__global__ void k_init_phase(float* __restrict__ xhat,
                             unsigned* __restrict__ m_ord,
                             float* __restrict__ den,
                             float* __restrict__ den2) {
    int i = blockIdx.x * blockDim.x + threadIdx.x;
    if (i < NW) xhat[i] = 0.0f;
    if (i < NN) { m_ord[i] = ORD_MIN; den[i] = 0.0f; den2[i] = 0.0f; }
}

// WMMA edge scoring: wave handles 16 edges; D = Xdst (16x64) * XsrcT (64x16),
// accumulated over NMAT modality matrices; edge scores are diag(D).
// NMAT templated so xp/yp/zp stay direct kernel args -> global_load (not flat).
template <int NMAT>
__global__ void k_scores_t(const float* __restrict__ xp,
                           const float* __restrict__ yp,
                           const float* __restrict__ zp,
                           const int* __restrict__ ei0,
                           float* __restrict__ sbuf) {
    const int ntiles = NE2 / 16;
    int tile = blockIdx.x * (blockDim.x >> 5) + (threadIdx.x >> 5);
    if (tile >= ntiles) return;
    int lane  = threadIdx.x & 31;
    int ebase = tile * 16;
    int e     = ebase + (lane & 15);
    int sN, dN;
    edge_sd(ei0, e, sN, dN);
    // f32 16x16x4 A layout: lanes 0-15 -> K = 4k+{0,1}; lanes 16-31 -> K = 4k+{2,3}
    unsigned koff = ((unsigned)lane >> 4) << 1;
    unsigned offA = (unsigned)dN * DIMC + koff;   // A rows = x[dst]
    unsigned offB = (unsigned)sN * DIMC + koff;   // B cols = x[src]

    v8f c = {0.f, 0.f, 0.f, 0.f, 0.f, 0.f, 0.f, 0.f};
    #pragma unroll
    for (int k = 0; k < 16; ++k) {
        {
            v2f a = *(const v2f*)(xp + offA + 4 * k);
            v2f b = *(const v2f*)(xp + offB + 4 * k);
            c = __builtin_amdgcn_wmma_f32_16x16x4_f32(
                    false, a, false, b, (short)0, c, false, false);
        }
        if (NMAT == 3) {
            v2f a = *(const v2f*)(yp + offA + 4 * k);
            v2f b = *(const v2f*)(yp + offB + 4 * k);
            c = __builtin_amdgcn_wmma_f32_16x16x4_f32(
                    false, a, false, b, (short)0, c, false, false);
            v2f a2 = *(const v2f*)(zp + offA + 4 * k);
            v2f b2 = *(const v2f*)(zp + offB + 4 * k);
            c = __builtin_amdgcn_wmma_f32_16x16x4_f32(
                    false, a2, false, b2, (short)0, c, false, false);
        }
    }
    // diag(D): element (e,e) at VGPR e&7, lane (e<8 ? e : e+16)
    int l7 = lane & 7;
    float dval = c[0];
    dval = (l7 == 1) ? c[1] : dval;
    dval = (l7 == 2) ? c[2] : dval;
    dval = (l7 == 3) ? c[3] : dval;
    dval = (l7 == 4) ? c[4] : dval;
    dval = (l7 == 5) ? c[5] : dval;
    dval = (l7 == 6) ? c[6] : dval;
    dval = (l7 == 7) ? c[7] : dval;
    int e16 = lane & 15;
    float sc = __shfl(dval, (e16 < 8) ? e16 : (e16 + 8), 32);
    if (lane < 16) {
        sc *= INV_SQRT_D;
        sc = (sc > 0.0f) ? sc : SLOPE * sc;     // leaky relu
        sbuf[ebase + lane] = sc;
    }
}

__global__ void k_segmax(const float* __restrict__ sbuf,
                         const int* __restrict__ ei0,
                         unsigned* __restrict__ m_ord) {
    int e = blockIdx.x * blockDim.x + threadIdx.x;
    if (e >= NE2) return;
    int sN, dN; edge_sd(ei0, e, sN, dN);
    atomicMax(&m_ord[dN], f2ord(sbuf[e]));
}

__global__ void k_expsum(float* __restrict__ sbuf,
                         const int* __restrict__ ei0,
                         const unsigned* __restrict__ m_ord,
                         float* __restrict__ den) {
    int e = blockIdx.x * blockDim.x + threadIdx.x;
    if (e >= NE2) return;
    int sN, dN; edge_sd(ei0, e, sN, dN);
    float ex = __expf(sbuf[e] - ord2f(m_ord[dN]));
    sbuf[e] = ex;
    atomicAdd(&den[dN], ex);
}

__global__ void k_alpha(float* __restrict__ sbuf,
                        const int* __restrict__ ei0,
                        const float* __restrict__ den,
                        float* __restrict__ den2,
                        int do_thresh) {
    int e = blockIdx.x * blockDim.x + threadIdx.x;
    if (e >= NE2) return;
    int sN, dN; edge_sd(ei0, e, sN, dN);
    float a = sbuf[e] / (den[dN] + EPS_F);
    if (do_thresh) {
        a = (a > THR_F) ? a : 0.0f;
        sbuf[e] = a;
        atomicAdd(&den2[dN], a);
    } else {
        sbuf[e] = a;
    }
}

// re-attention renormalize; also emits alpha_t into d_out tail
__global__ void k_renorm(float* __restrict__ sbuf,
                         const int* __restrict__ ei0,
                         const float* __restrict__ den2,
                         float* __restrict__ alpha_out) {
    int e = blockIdx.x * blockDim.x + threadIdx.x;
    if (e >= NE2) return;
    int sN, dN; edge_sd(ei0, e, sN, dN);
    float a = sbuf[e] / (den2[dN] + EPS_F);
    sbuf[e] = a;
    alpha_out[e] = a;
}

// x_hat[dst] += alpha * x[src] ; one thread per (edge, 4 components)
__global__ void k_aggregate(const float* __restrict__ x,
                            const float* __restrict__ sbuf,
                            const int* __restrict__ ei0,
                            float* __restrict__ xhat) {
    int t = blockIdx.x * blockDim.x + threadIdx.x;
    if (t >= NE2 * 16) return;
    int e = t >> 4;
    unsigned q = ((unsigned)t & 15u) << 2;
    float a = sbuf[e];
    if (a == 0.0f) return;          // thresholded edges contribute nothing
    int sN, dN; edge_sd(ei0, e, sN, dN);
    float4 v = *(const float4*)(x + (unsigned)sN * DIMC + q);
    float* o = xhat + (unsigned)dN * DIMC + q;
    atomicAdd(o + 0, a * v.x);
    atomicAdd(o + 1, a * v.y);
    atomicAdd(o + 2, a * v.z);
    atomicAdd(o + 3, a * v.w);
}

// x = l2norm(x + x_hat); one wave per row
__global__ void k_addnorm(float* __restrict__ x,
                          const float* __restrict__ xhat) {
    int row = blockIdx.x * (blockDim.x >> 5) + (threadIdx.x >> 5);
    if (row >= NN) return;
    int lane = threadIdx.x & 31;
    unsigned b = (unsigned)row * DIMC + lane * 2;
    float2 v = *(const float2*)(x + b);
    float2 h = *(const float2*)(xhat + b);
    float2 o = l2pair(v.x + h.x, v.y + h.y);
    *(float2*)(x + b) = o;
}

// ---------------- host launcher ----------------

static void run_pass(float* x, int nmat, const float* y, const float* z,
                     bool thresh, float* alpha_out,
                     float* xhat, float* sbuf, unsigned* m_ord,
                     float* den, float* den2,
                     const int* ei0, hipStream_t stream) {
    const int T = 256;
    k_init_phase<<<(NW + T - 1) / T, T, 0, stream>>>(xhat, m_ord, den, den2);
    if (nmat == 3)
        k_scores_t<3><<<((NE2 / 16) + 7) / 8, T, 0, stream>>>(x, y, z, ei0, sbuf);
    else
        k_scores_t<1><<<((NE2 / 16) + 7) / 8, T, 0, stream>>>(x, x, x, ei0, sbuf);
    k_segmax<<<(NE2 + T - 1) / T, T, 0, stream>>>(sbuf, ei0, m_ord);
    k_expsum<<<(NE2 + T - 1) / T, T, 0, stream>>>(sbuf, ei0, m_ord, den);
    k_alpha<<<(NE2 + T - 1) / T, T, 0, stream>>>(sbuf, ei0, den, den2,
                                                 thresh ? 1 : 0);
    if (thresh)
        k_renorm<<<(NE2 + T - 1) / T, T, 0, stream>>>(sbuf, ei0, den2, alpha_out);
    k_aggregate<<<(NE2 * 16) / T, T, 0, stream>>>(x, sbuf, ei0, xhat);
    k_addnorm<<<(NN + 7) / 8, T, 0, stream>>>(x, xhat);
}

extern "C" void kernel_launch(void* const* d_in, const int* in_sizes, int n_in,
                              void* d_out, int out_size, void* d_ws, size_t ws_size,
                              hipStream_t stream) {
    const float* features  = (const float*)d_in[0];
    const float* features2 = (const float*)d_in[1];
    const float* features3 = (const float*)d_in[2];
    const float* preference= (const float*)d_in[3];
    const int*   ei0       = (const int*)d_in[4];

    float* x         = (float*)d_out;        // working embeddings live in d_out head
    float* alpha_out = x + NW;               // alpha_t tail

    float* y    = (float*)d_ws;
    float* z    = y + NW;
    float* xhat = z + NW;
    float* sbuf = xhat + NW;
    unsigned* m_ord = (unsigned*)(sbuf + NE2);
    float* den  = (float*)(m_ord + NN);
    float* den2 = den + NN;

    const int T = 256;
    k_init_nodes<<<(NN + 7) / 8, T, 0, stream>>>(features, features2, features3,
                                                 preference, x, y, z);

    // conv_embed_1: 3-modality scores, threshold + re-attention
    run_pass(x, 3, y, z, true, alpha_out, xhat, sbuf, m_ord, den, den2, ei0, stream);
    // 3 routing iterations: plain dot-product GAT on x
    for (int r = 0; r < 3; ++r)
        run_pass(x, 1, nullptr, nullptr, false, nullptr,
                 xhat, sbuf, m_ord, den, den2, ei0, stream);
}